// GaugeAttention_77687368450655
// MI455X (gfx1250) — compile-verified
//
#include <hip/hip_runtime.h>
#include <hip/hip_bf16.h>
#include <math.h>

#define NN 128
#define GG 16
#define KK 16

typedef __attribute__((ext_vector_type(2))) float v2f;
typedef __attribute__((ext_vector_type(8))) float v8f;

// ---------------------------------------------------------------------------
// Wave-cooperative in-place Cholesky of a 16x16 SPD matrix in LDS.
// Lower triangle (incl. diagonal) holds L afterwards; upper triangle is stale.
// blockDim.x == 32 (one wave).
// ---------------------------------------------------------------------------
__device__ __forceinline__ void chol16(float* S, int lane) {
#pragma unroll 1
  for (int k = 0; k < 16; ++k) {
    float d = sqrtf(S[k * 16 + k]);   // broadcast read (all lanes same)
    float inv = 1.0f / d;
    __syncthreads();
    if (lane == k) S[k * 16 + k] = d;
    if (lane > k && lane < 16) S[lane * 16 + k] *= inv;
    __syncthreads();
#pragma unroll
    for (int q = 0; q < 8; ++q) {
      int e = lane + 32 * q;
      int r = e >> 4, c = e & 15;
      if (r > k && c > k && c <= r) S[r * 16 + c] -= S[r * 16 + k] * S[c * 16 + k];
    }
    __syncthreads();
  }
}

__device__ __forceinline__ float wsum(float v) {
#pragma unroll
  for (int s = 16; s > 0; s >>= 1) v += __shfl_xor(v, s, 32);
  return v;
}
__device__ __forceinline__ float wmax(float v) {
#pragma unroll
  for (int s = 16; s > 0; s >>= 1) v = fmaxf(v, __shfl_xor(v, s, 32));
  return v;
}

// ---------------------------------------------------------------------------
// Phase 0: per (n,g) Cholesky of sigma_q -> L1 (upper zeroed) + logdet1.
// grid = N*G blocks of 32.
// ---------------------------------------------------------------------------
__global__ void k_prep(const float* __restrict__ sigma_q,
                       float* __restrict__ wsL1,
                       float* __restrict__ wsLd) {
  __shared__ float sS[256];
  int lane = threadIdx.x;
  int id = blockIdx.x;  // n*G + g
  const float4* src = (const float4*)(sigma_q + (size_t)id * 256);
  ((float4*)sS)[lane * 2 + 0] = src[lane * 2 + 0];
  ((float4*)sS)[lane * 2 + 1] = src[lane * 2 + 1];
  __syncthreads();
  chol16(sS, lane);
  float* outp = wsL1 + (size_t)id * 256;
#pragma unroll
  for (int q = 0; q < 8; ++q) {
    int e = lane * 8 + q;
    int r = e >> 4, c = e & 15;
    outp[e] = (c <= r) ? sS[e] : 0.0f;
  }
  if (lane == 0) {
    float ld = 0.0f;
#pragma unroll
    for (int r = 0; r < 16; ++r) ld += logf(sS[r * 17]);
    wsLd[id] = 2.0f * ld;
  }
}

// ---------------------------------------------------------------------------
// Phase 1: per (i,j,g) tile, one wave. WMMA congruence + Cholesky + solves.
// grid = N*N*G blocks of 32. Writes E_ij into out[EOFF + idx].
// ---------------------------------------------------------------------------
__global__ void k_energy(const float* __restrict__ mu_q,
                         const float* __restrict__ sigma_q,
                         const float* __restrict__ omegas,
                         const float* __restrict__ wsL1,
                         const float* __restrict__ wsLd,
                         float* __restrict__ out) {
  __shared__ float sO[256];   // Omega_ijg
  __shared__ float sS[256];   // Sigma_j, then Sigma_t / L2 (in-place chol)
  __shared__ float sM[256];   // M1 = Omega*Sigma, then L1_i
  __shared__ float sMu[16];   // mu_j
  __shared__ float sD[16];    // diff = Omega*mu_j - mu_i
  __shared__ float sP[32];    // per-lane partial sums

  const int EOFF = NN * NN * GG;
  int lane = threadIdx.x;
  int idx = blockIdx.x;                 // (i*N + j)*G + g  == E linear index
  int g = idx % GG;
  int j = (idx / GG) % NN;
  int i = idx / (GG * NN);

  {
    const float4* Op = (const float4*)(omegas + (size_t)idx * 256);
    const float4* Sp = (const float4*)(sigma_q + (size_t)(j * GG + g) * 256);
    ((float4*)sO)[lane * 2 + 0] = Op[lane * 2 + 0];
    ((float4*)sO)[lane * 2 + 1] = Op[lane * 2 + 1];
    ((float4*)sS)[lane * 2 + 0] = Sp[lane * 2 + 0];
    ((float4*)sS)[lane * 2 + 1] = Sp[lane * 2 + 1];
    if (lane < 16) sMu[lane] = mu_q[(size_t)(j * GG + g) * 16 + lane];
  }
  __syncthreads();

  int m = lane & 15;    // A-row / B-col / C-col index
  int h = lane >> 4;    // lane half

  // ---- M1 = Omega * Sigma_j : 4 accumulating V_WMMA_F32_16X16X4_F32 ----
  v8f acc = {0.f, 0.f, 0.f, 0.f, 0.f, 0.f, 0.f, 0.f};
#pragma unroll
  for (int kc = 0; kc < 16; kc += 4) {
    v2f a, b;
    a[0] = sO[m * 16 + kc + 2 * h + 0];          // A[m][K], K = v + 2h + kc
    a[1] = sO[m * 16 + kc + 2 * h + 1];
    b[0] = sS[(kc + 2 * h + 0) * 16 + m];        // B[K][n], n = m
    b[1] = sS[(kc + 2 * h + 1) * 16 + m];
    acc = __builtin_amdgcn_wmma_f32_16x16x4_f32(false, a, false, b,
                                                (short)0, acc, false, false);
  }
#pragma unroll
  for (int r = 0; r < 8; ++r) sM[(r + 8 * h) * 16 + m] = acc[r];  // C: row r+8h
  __syncthreads();

  // ---- Sigma_t = M1 * Omega^T : 4 more WMMA, B read transposed from sO ----
  v8f acc2 = {0.f, 0.f, 0.f, 0.f, 0.f, 0.f, 0.f, 0.f};
#pragma unroll
  for (int kc = 0; kc < 16; kc += 4) {
    v2f a, b;
    a[0] = sM[m * 16 + kc + 2 * h + 0];
    a[1] = sM[m * 16 + kc + 2 * h + 1];
    b[0] = sO[m * 16 + kc + 2 * h + 0];          // Omega^T[K][n] = Omega[n][K]
    b[1] = sO[m * 16 + kc + 2 * h + 1];
    acc2 = __builtin_amdgcn_wmma_f32_16x16x4_f32(false, a, false, b,
                                                 (short)0, acc2, false, false);
  }
  __syncthreads();
#pragma unroll
  for (int r = 0; r < 8; ++r) sS[(r + 8 * h) * 16 + m] = acc2[r];  // overwrite Sigma_j

  // ---- diff = Omega * mu_j - mu_i ----
  if (lane < 16) {
    float s = 0.0f;
#pragma unroll
    for (int k = 0; k < 16; ++k) s += sO[lane * 16 + k] * sMu[k];
    sD[lane] = s - mu_q[(size_t)(i * GG + g) * 16 + lane];
  }
  __syncthreads();

  // ---- L2 = chol(Sigma_t) in-place ----
  chol16(sS, lane);

  // ---- load L1_i ----
  {
    const float4* Lp = (const float4*)(wsL1 + (size_t)(i * GG + g) * 256);
    ((float4*)sM)[lane * 2 + 0] = Lp[lane * 2 + 0];
    ((float4*)sM)[lane * 2 + 1] = Lp[lane * 2 + 1];
  }
  __syncthreads();

  // ---- forward substitution: lanes 0..15 solve L2 X = L1 (col = lane),
  //      lane 16 solves L2 v = diff. Accumulate sum of squares. ----
  float xs[16];
  float part = 0.0f;
#pragma unroll
  for (int r = 0; r < 16; ++r) {
    float brc = (lane < 16) ? sM[r * 16 + lane]
                            : ((lane == 16) ? sD[r] : 0.0f);
    float x = brc;
#pragma unroll
    for (int t = 0; t < r; ++t) x -= sS[r * 16 + t] * xs[t];
    x /= sS[r * 16 + r];
    xs[r] = x;
    part += x * x;
  }
  sP[lane] = part;
  __syncthreads();

  if (lane == 0) {
    float tr = 0.0f;
#pragma unroll
    for (int c = 0; c < 16; ++c) tr += sP[c];
    float maha = sP[16];
    float ld2 = 0.0f;
#pragma unroll
    for (int r = 0; r < 16; ++r) ld2 += logf(sS[r * 17]);
    ld2 *= 2.0f;
    float E = 0.5f * (tr + maha - 16.0f + ld2 - wsLd[(size_t)(i * GG + g)]);
    out[EOFF + (size_t)idx] = E;
  }
}

// ---------------------------------------------------------------------------
// Phase 2: per (i,g) masked softmax over j, beta + entropy + F partial.
// grid = N*G blocks of 32; each lane handles 4 j's.
// ---------------------------------------------------------------------------
__global__ void k_soft(float* __restrict__ out, float* __restrict__ wsF) {
  const int EOFF = NN * NN * GG;
  const int ENTOFF = 2 * NN * NN * GG + 1;
  int lane = threadIdx.x;
  int blk = blockIdx.x;  // i*G + g
  int g = blk % GG;
  int i = blk / GG;

  float e[4], z[4];
  float mx = -3.0e38f;
#pragma unroll
  for (int q = 0; q < 4; ++q) {
    int jj = lane + 32 * q;
    e[q] = out[EOFF + (size_t)((i * NN + jj) * GG + g)];
    z[q] = (jj == i) ? -1.0e30f : -e[q];   // TEMP = 1, self masked
    mx = fmaxf(mx, z[q]);
  }
  mx = wmax(mx);

  float p[4];
  float sum = 0.0f;
#pragma unroll
  for (int q = 0; q < 4; ++q) {
    int jj = lane + 32 * q;
    p[q] = (jj == i) ? 0.0f : expf(z[q] - mx);
    sum += p[q];
  }
  sum = wsum(sum);
  float inv = 1.0f / sum;

  float ent = 0.0f, fbe = 0.0f;
#pragma unroll
  for (int q = 0; q < 4; ++q) {
    int jj = lane + 32 * q;
    float beta = p[q] * inv;
    out[(size_t)((i * NN + jj) * GG + g)] = beta;
    ent -= beta * logf(beta + 1e-10f);
    fbe += beta * e[q];
  }
  ent = wsum(ent);
  fbe = wsum(fbe);
  if (lane == 0) {
    out[ENTOFF + (size_t)blk] = ent;
    wsF[blk] = fbe - ent;
  }
}

// ---------------------------------------------------------------------------
// Phase 3: deterministic fixed-order reduction of 2048 F partials.
// ---------------------------------------------------------------------------
__global__ void k_fred(const float* __restrict__ wsF, float* __restrict__ out) {
  const int FOFF = 2 * NN * NN * GG;
  __shared__ float sh[256];
  int t = threadIdx.x;
  float s = 0.0f;
  for (int k = t; k < NN * GG; k += 256) s += wsF[k];
  sh[t] = s;
  __syncthreads();
  for (int st = 128; st > 0; st >>= 1) {
    if (t < st) sh[t] += sh[t + st];
    __syncthreads();
  }
  if (t == 0) out[FOFF] = sh[0];
}

// ---------------------------------------------------------------------------
extern "C" void kernel_launch(void* const* d_in, const int* in_sizes, int n_in,
                              void* d_out, int out_size, void* d_ws, size_t ws_size,
                              hipStream_t stream) {
  const float* mu_q    = (const float*)d_in[0];
  const float* sigma_q = (const float*)d_in[1];
  const float* omegas  = (const float*)d_in[2];
  float* out = (float*)d_out;

  float* wsL1 = (float*)d_ws;                     // N*G*256 floats (2 MB)
  float* wsLd = wsL1 + (size_t)NN * GG * 256;     // N*G floats
  float* wsF  = wsLd + (size_t)NN * GG;           // N*G floats

  k_prep<<<NN * GG, 32, 0, stream>>>(sigma_q, wsL1, wsLd);
  k_energy<<<NN * NN * GG, 32, 0, stream>>>(mu_q, sigma_q, omegas, wsL1, wsLd, out);
  k_soft<<<NN * GG, 32, 0, stream>>>(out, wsF);
  k_fred<<<1, 256, 0, stream>>>(wsF, out);
}